// Fishnet_15710990369303
// MI455X (gfx1250) — compile-verified
//
#include <hip/hip_runtime.h>
#include <math.h>

typedef __attribute__((ext_vector_type(2))) float v2f;
typedef __attribute__((ext_vector_type(4))) float v4f;
typedef __attribute__((ext_vector_type(8))) float v8f;

#define NETOUT 152   // 16 (t) + 136 (tril of 16x16)
#define NCT 10       // 160 padded columns / 16
#define NWAVES 4     // waves per block in the fused kernel

// Branch-free tanh: clamp so exp stays finite; saturation is exact in fp32.
__device__ __forceinline__ float fast_tanh(float x) {
    x = fminf(fmaxf(x, -15.0f), 15.0f);
    const float e = __expf(2.0f * x);                    // native v_exp_f32 path
    return (e - 1.0f) * __builtin_amdgcn_rcpf(e + 1.0f); // single trans rcp
}
// Branch-free softplus for x in (-1,1): log argument in (1.37, 3.72).
__device__ __forceinline__ float fast_softplus(float x) {
    return __logf(1.0f + __expf(x));
}

// ---------------------------------------------------------------------------
// Kernel 0: initialize reduction buffers. F_red starts as identity (this
// absorbs the "+ eye(16)" of the reference), t_red starts at zero.
// ---------------------------------------------------------------------------
__global__ void init_red_kernel(float* __restrict__ Fred, float* __restrict__ tred,
                                int nseg) {
    const size_t stride = (size_t)gridDim.x * blockDim.x;
    const size_t tid0   = (size_t)blockIdx.x * blockDim.x + threadIdx.x;
    const size_t ftotal = (size_t)nseg * 256;
    for (size_t i = tid0; i < ftotal; i += stride) {
        // within each 16x16 block, diagonal entries are at offsets 17*i
        Fred[i] = (((unsigned)(i & 255u) % 17u) == 0u) ? 1.0f : 0.0f;
    }
    const size_t ttotal = (size_t)nseg * 16;
    for (size_t i = tid0; i < ttotal; i += stride) tred[i] = 0.0f;
}

// ---------------------------------------------------------------------------
// Kernel 1: fused  net_out = data@W+b  ->  L (tanh/softplus)  ->  F = L L^T
//           -> atomic segment scatter of F (lower triangle) and t.
// One wave per 16-row tile. All matrix math via V_WMMA_F32_16X16X4_F32.
// ---------------------------------------------------------------------------
__global__ void fused_main_kernel(const float* __restrict__ data,
                                  const int*   __restrict__ segments,
                                  const float* __restrict__ W,     // [64 x 152]
                                  const float* __restrict__ bvec,  // [152]
                                  float* __restrict__ Fred,        // [nseg x 16 x 16]
                                  float* __restrict__ tred,        // [nseg x 16]
                                  int ntiles) {
    __shared__ float Lbuf[NWAVES * 16 * 16 * 16];   // per-wave 16 L-matrices (16x16)

    const int tid  = threadIdx.x;
    const int wave = tid >> 5;
    const int lane = tid & 31;
    const int h    = lane >> 4;    // 0: K even pair, 1: K odd pair (A/B layout halves)
    const int m    = lane & 15;    // M row (A) == N col (B/C/D) within tile

    const int tile = blockIdx.x * NWAVES + wave;
    if (tile >= ntiles) return;    // grid sized exactly; whole-wave uniform anyway
    const size_t rows_base = (size_t)tile * 16;

    // ---- Load A tile (16 rows x 64) into WMMA A-layout: per lane 16 float2 ----
    // A 16x4 f32 layout: lanes 0-15 hold K={0,1}, lanes 16-31 hold K={2,3}.
    v2f a[16];
    const float* dptr = data + (rows_base + m) * 64 + 2 * h;
    #pragma unroll
    for (int k = 0; k < 16; ++k)
        a[k] = *(const v2f*)(dptr + 4 * k);

    // ---- Zero this wave's L staging area (wave-private: no barrier needed) ----
    float* Lw = Lbuf + wave * 4096;
    {
        v4f z = {0.0f, 0.0f, 0.0f, 0.0f};
        #pragma unroll
        for (int i = 0; i < 32; ++i)
            *(v4f*)(Lw + (i * 32 + lane) * 4) = z;
    }

    // ---- GEMM over 10 column tiles; transform; stash t / scatter L ----
    float tvals[8];
    for (int ct = 0; ct < NCT; ++ct) {
        const int col   = ct * 16 + m;
        const bool valid = (col < NETOUT);
        // Clamped, UNCONDITIONAL B loads: garbage only reaches D columns we
        // discard (B column j only feeds D column j, owned by lanes m==j).
        const int colc  = valid ? col : (NETOUT - 1);

        // Batch-preload all B operands for this column tile, then run the
        // 16-deep WMMA chain back-to-back (loads overlap the chain).
        v2f breg[16];
        #pragma unroll
        for (int k = 0; k < 16; ++k) {
            // B 4x16 f32 layout mirrors A: lanes 0-15 K={0,1}, lanes 16-31 K={2,3}
            const int krow = 4 * k + 2 * h;
            breg[k][0] = W[(size_t)krow * NETOUT + colc];
            breg[k][1] = W[(size_t)(krow + 1) * NETOUT + colc];
        }

        v8f acc = {0.f, 0.f, 0.f, 0.f, 0.f, 0.f, 0.f, 0.f};
        #pragma unroll
        for (int k = 0; k < 16; ++k) {
            acc = __builtin_amdgcn_wmma_f32_16x16x4_f32(
                      false, a[k], false, breg[k], (short)0, acc, false, false);
        }
        const float bb = bvec[colc];

        if (ct == 0) {
            // t columns: keep for atomic scatter (row = v + 8*h in C layout)
            #pragma unroll
            for (int v = 0; v < 8; ++v) tvals[v] = acc[v] + bb;
        } else {
            // y columns: unrank triangular index q -> (i,j), tanh (+softplus on diag)
            const int q = col - 16;                       // 0..143 (>=136 invalid)
            int i = (int)floorf((sqrtf(8.0f * (float)q + 1.0f) - 1.0f) * 0.5f);
            if ((i + 1) * (i + 2) / 2 <= q) ++i;
            if (i * (i + 1) / 2 > q) --i;
            const int  j    = q - i * (i + 1) / 2;
            const bool diag = (i == j);
            #pragma unroll
            for (int v = 0; v < 8; ++v) {
                float val = fast_tanh(acc[v] + bb);
                const float sp = fast_softplus(val);      // softplus(tanh(.))
                val = diag ? sp : val;                    // v_cndmask, no branch
                if (valid) Lw[(v + 8 * h) * 256 + i * 16 + j] = val;
            }
        }
    }

    // ---- scatter t into t_red ----
    #pragma unroll
    for (int v = 0; v < 8; ++v) {
        const int r = v + 8 * h;
        const int s = segments[rows_base + r];
        unsafeAtomicAdd(&tred[(size_t)s * 16 + m], tvals[v]);
    }

    // ---- F = L L^T per row via WMMA; scatter lower triangle into F_red ----
    for (int r = 0; r < 16; ++r) {
        const float* Lr = Lw + r * 256;
        // A needs L[m][k..], B needs (L^T)[k..][n] = L[n][k..]; n==m per lane
        // -> identical operand, one 8B LDS load feeds both. Preload all 4.
        v2f la[4];
        #pragma unroll
        for (int kk = 0; kk < 4; ++kk)
            la[kk] = *(const v2f*)(Lr + m * 16 + 4 * kk + 2 * h);

        v8f f = {0.f, 0.f, 0.f, 0.f, 0.f, 0.f, 0.f, 0.f};
        #pragma unroll
        for (int kk = 0; kk < 4; ++kk) {
            f = __builtin_amdgcn_wmma_f32_16x16x4_f32(
                    false, la[kk], false, la[kk], (short)0, f, false, false);
        }
        const int s = segments[rows_base + r];
        const size_t base = (size_t)s * 256;
        #pragma unroll
        for (int v = 0; v < 8; ++v) {
            const int i = v + 8 * h;                 // row of F, col = m
            if (i >= m)                              // symmetric: lower triangle only
                unsafeAtomicAdd(&Fred[base + i * 16 + m], f[v]);
        }
    }
}

// ---------------------------------------------------------------------------
// Kernel 2: per segment, solve (F_red) x = 1 (SPD, no pivoting), then
// theta = t_red * x  (== colsum(F^-1) * t_red) and out = theta @ proj_W.
// One 16-lane group per segment, row-per-lane Gaussian elimination.
// ---------------------------------------------------------------------------
__global__ void solve_project_kernel(const float* __restrict__ Fred,
                                     const float* __restrict__ tred,
                                     const float* __restrict__ projW, // [16 x 64]
                                     float* __restrict__ out,         // [nseg x 64]
                                     int nseg) {
    const int tid = threadIdx.x;
    const int grp = tid >> 4;
    const int r   = tid & 15;
    const int s   = blockIdx.x * (blockDim.x >> 4) + grp;
    if (s >= nseg) return;

    const float* Fs = Fred + (size_t)s * 256;
    float arr[16];
    #pragma unroll
    for (int j = 0; j < 16; ++j)   // mirror: only lower triangle was accumulated
        arr[j] = (j <= r) ? Fs[r * 16 + j] : Fs[j * 16 + r];
    float rhs = 1.0f;

    // Forward elimination
    #pragma unroll
    for (int k = 0; k < 16; ++k) {
        float pk[16];
        #pragma unroll
        for (int j = 0; j < 16; ++j) pk[j] = __shfl(arr[j], k, 16);
        const float prhs = __shfl(rhs, k, 16);
        if (r > k) {
            const float fct = arr[k] / pk[k];
            #pragma unroll
            for (int j = 0; j < 16; ++j)
                if (j >= k) arr[j] -= fct * pk[j];
            rhs -= fct * prhs;
        }
    }

    // Back substitution
    float x = 0.0f;
    #pragma unroll
    for (int k = 15; k >= 0; --k) {
        const float num = __shfl(rhs, k, 16);
        const float dpv = __shfl(arr[k], k, 16);   // lane k's diagonal
        const float xk  = num / dpv;
        if (r == k) x = xk;
        if (r < k)  rhs -= arr[k] * xk;
    }

    const float theta = tred[(size_t)s * 16 + r] * x;

    // out[s, :] = theta[16] @ proj_W[16x64]; lane r covers cols r, r+16, r+32, r+48
    float o0 = 0.f, o1 = 0.f, o2 = 0.f, o3 = 0.f;
    #pragma unroll
    for (int j = 0; j < 16; ++j) {
        const float th = __shfl(theta, j, 16);
        o0 += th * projW[j * 64 + r];
        o1 += th * projW[j * 64 + r + 16];
        o2 += th * projW[j * 64 + r + 32];
        o3 += th * projW[j * 64 + r + 48];
    }
    float* op = out + (size_t)s * 64;
    op[r]      = o0;
    op[r + 16] = o1;
    op[r + 32] = o2;
    op[r + 48] = o3;
}

// ---------------------------------------------------------------------------
extern "C" void kernel_launch(void* const* d_in, const int* in_sizes, int n_in,
                              void* d_out, int out_size, void* d_ws, size_t ws_size,
                              hipStream_t stream) {
    const float* data     = (const float*)d_in[0];
    const int*   segments = (const int*)d_in[1];
    // d_in[2] = num_segments (device scalar); derive sizes on host instead:
    const float* W        = (const float*)d_in[3];
    const float* bvec     = (const float*)d_in[4];
    const float* projW    = (const float*)d_in[5];

    const int N    = in_sizes[0] / 64;   // data is [N, 64]
    const int nseg = out_size / 64;      // output is [nseg, 64]

    float* Fred = (float*)d_ws;                       // nseg * 256 floats
    float* tred = Fred + (size_t)nseg * 256;          // nseg * 16 floats

    // 0) init reduction buffers (identity in F_red)
    init_red_kernel<<<4096, 256, 0, stream>>>(Fred, tred, nseg);

    // 1) fused GEMM + L + F + scatter (one wave per 16-row tile)
    const int ntiles = (N + 15) / 16;
    const int blocks = (ntiles + NWAVES - 1) / NWAVES;
    fused_main_kernel<<<blocks, NWAVES * 32, 0, stream>>>(
        data, segments, W, bvec, Fred, tred, ntiles);

    // 2) per-segment solve + projection (16 segments per 256-thread block)
    const int segs_per_block = 256 / 16;
    const int sblocks = (nseg + segs_per_block - 1) / segs_per_block;
    solve_project_kernel<<<sblocks, 256, 0, stream>>>(Fred, tred, projW,
                                                      (float*)d_out, nseg);
}